// CTPNLoss_17660905521640
// MI455X (gfx1250) — compile-verified
//
#include <hip/hip_runtime.h>

typedef __attribute__((ext_vector_type(2))) float v2f;
typedef __attribute__((ext_vector_type(8))) float v8f;

#define CTPN_W   1024
#define CTPN_HW  (512 * 1024)

__device__ __forceinline__ float smooth_l1(float d) {
    float ad = fabsf(d);
    return (ad < 1.0f) ? (0.5f * d * d) : (ad - 0.5f);
}

// Fold the 8-VGPR f32 accumulator of a 16x16 WMMA D-matrix (with B = ones,
// so every column is identical) into a single scalar.
// Column 0: lane 0 holds M=0..7 in c[0..7], lane 16 holds M=8..15.
__device__ __forceinline__ float wmma_acc_to_scalar(v8f c) {
    float s = c[0] + c[1] + c[2] + c[3] + c[4] + c[5] + c[6] + c[7];
    return __shfl(s, 0, 32) + __shfl(s, 16, 32);
}

// Sum 64 consecutive f32 values (mapped into the 16x4 A layout) into the
// carried accumulator via V_WMMA_F32_16X16X4_F32 (exact f32 FMA path).
// A layout (32 lanes x 2 VGPRs): lanes 0-15 -> (M=lane, K=0/1),
// lanes 16-31 -> (M=lane-16, K=2/3). Index p = M*4 + K is a bijection onto 0..63.
__device__ __forceinline__ v8f wmma_sum64(v8f c, float a0, float a1) {
    v2f a;  a.x = a0;  a.y = a1;
    v2f b;  b.x = 1.0f; b.y = 1.0f;           // B = ones(4x16)
    return __builtin_amdgcn_wmma_f32_16x16x4_f32(
        /*neg_a=*/false, a, /*neg_b=*/false, b,
        /*c_mod=*/(short)0, c, /*reuse_a=*/false, /*reuse_b=*/false);
}

__global__ void __launch_bounds__(256)
ctpn_partial_kernel(const float* __restrict__ score,
                    const float* __restrict__ vpred,
                    const float* __restrict__ spred,
                    const int* __restrict__ pos_j, const int* __restrict__ pos_i, const int* __restrict__ pos_k,
                    const int* __restrict__ neg_j, const int* __restrict__ neg_i, const int* __restrict__ neg_k,
                    const int* __restrict__ vr_j,  const int* __restrict__ vr_i,  const int* __restrict__ vr_k,
                    const float* __restrict__ vr_tgt,
                    const int* __restrict__ sr_j,  const int* __restrict__ sr_i,  const int* __restrict__ sr_k,
                    const float* __restrict__ sr_tgt,
                    float* __restrict__ partials,
                    int Np, int Nn, int Nv, int Ns,
                    float w_cls, float w_v, float w_s) {
    __shared__ float sm[256];

    const int tid = threadIdx.x;
    const int t   = blockIdx.x * 256 + tid;
    const int total = Np + Nn + Nv + Ns;

    float contrib = 0.0f;
    if (t < total) {
        if (t < Np + Nn) {
            // classification: 2-way cross entropy
            bool is_pos = (t < Np);
            int u = is_pos ? t : (t - Np);
            int j = is_pos ? pos_j[u] : neg_j[u];
            int i = is_pos ? pos_i[u] : neg_i[u];
            int k = is_pos ? pos_k[u] : neg_k[u];
            long base = (long)(2 * k) * CTPN_HW + (long)j * CTPN_W + i;
            float l0 = score[base];
            float l1 = score[base + CTPN_HW];
            float m  = fmaxf(l0, l1);
            float lse = m + logf(expf(l0 - m) + expf(l1 - m));
            float ce  = lse - (is_pos ? l1 : l0);
            contrib = ce * w_cls;
        } else if (t < Np + Nn + Nv) {
            // vertical regression: SmoothL1 on (vc, vh)
            int u = t - Np - Nn;
            int j = vr_j[u], i = vr_i[u], k = vr_k[u];
            long base = (long)(2 * k) * CTPN_HW + (long)j * CTPN_W + i;
            float p0 = vpred[base];
            float p1 = vpred[base + CTPN_HW];
            float d0 = p0 - vr_tgt[2 * u];
            float d1 = p1 - vr_tgt[2 * u + 1];
            contrib = (smooth_l1(d0) + smooth_l1(d1)) * w_v;
        } else {
            // side refinement: SmoothL1 on single offset
            int u = t - Np - Nn - Nv;
            int j = sr_j[u], i = sr_i[u], k = sr_k[u];
            long base = (long)k * CTPN_HW + (long)j * CTPN_W + i;
            float d = spred[base] - sr_tgt[u];
            contrib = smooth_l1(d) * w_s;
        }
    }

    sm[tid] = contrib;
    __syncthreads();

    // Wave 0 reduces all 256 contributions with 4 chained f32 WMMAs.
    if (tid < 32) {
        const int lane = tid;
        v8f c = {};
        #pragma unroll
        for (int ch = 0; ch < 4; ++ch) {
            int base = ch * 64;
            int i0 = (lane < 16) ? (base + 4 * lane) : (base + 4 * (lane - 16) + 2);
            c = wmma_sum64(c, sm[i0], sm[i0 + 1]);
        }
        float tot = wmma_acc_to_scalar(c);
        if (lane == 0) partials[blockIdx.x] = tot;
    }
}

__global__ void ctpn_final_kernel(const float* __restrict__ partials, int n,
                                  float* __restrict__ out) {
    const int lane = threadIdx.x;   // launched with exactly 32 threads (one wave)
    v8f c = {};
    const int chunks = (n + 63) >> 6;
    for (int ch = 0; ch < chunks; ++ch) {
        int base = ch * 64;
        int i0 = (lane < 16) ? (base + 4 * lane) : (base + 4 * (lane - 16) + 2);
        float a0 = (i0 < n)     ? partials[i0]     : 0.0f;
        float a1 = (i0 + 1 < n) ? partials[i0 + 1] : 0.0f;
        c = wmma_sum64(c, a0, a1);
    }
    float tot = wmma_acc_to_scalar(c);
    if (lane == 0) out[0] = tot;
}

extern "C" void kernel_launch(void* const* d_in, const int* in_sizes, int n_in,
                              void* d_out, int out_size, void* d_ws, size_t ws_size,
                              hipStream_t stream) {
    const float* score  = (const float*)d_in[0];
    const float* vpred  = (const float*)d_in[1];
    const float* spred  = (const float*)d_in[2];
    const int* pos_j = (const int*)d_in[3];
    const int* pos_i = (const int*)d_in[4];
    const int* pos_k = (const int*)d_in[5];
    const int* neg_j = (const int*)d_in[6];
    const int* neg_i = (const int*)d_in[7];
    const int* neg_k = (const int*)d_in[8];
    const int* vr_j  = (const int*)d_in[9];
    const int* vr_i  = (const int*)d_in[10];
    const int* vr_k  = (const int*)d_in[11];
    const float* vr_tgt = (const float*)d_in[12];
    const int* sr_j  = (const int*)d_in[13];
    const int* sr_i  = (const int*)d_in[14];
    const int* sr_k  = (const int*)d_in[15];
    const float* sr_tgt = (const float*)d_in[16];

    const int Np = in_sizes[3];
    const int Nn = in_sizes[6];
    const int Nv = in_sizes[9];
    const int Ns = in_sizes[13];
    const int total  = Np + Nn + Nv + Ns;
    const int blocks = (total + 255) / 256;

    const float w_cls = 1.0f / (float)(Np + Nn);
    const float w_v   = 1.0f / (float)(2 * Nv);
    const float w_s   = 1.0f / (float)Ns;

    float* partials = (float*)d_ws;   // blocks * 4 bytes (112 floats by default)

    ctpn_partial_kernel<<<blocks, 256, 0, stream>>>(
        score, vpred, spred,
        pos_j, pos_i, pos_k, neg_j, neg_i, neg_k,
        vr_j, vr_i, vr_k, vr_tgt, sr_j, sr_i, sr_k, sr_tgt,
        partials, Np, Nn, Nv, Ns, w_cls, w_v, w_s);

    ctpn_final_kernel<<<1, 32, 0, stream>>>(partials, blocks, (float*)d_out);
}